// GraphSAGE_Embs_58205396795521
// MI455X (gfx1250) — compile-verified
//
#include <hip/hip_runtime.h>
#include <hip/hip_bf16.h>

#define HC 128          // hidden channels
#define LDSP 132        // padded LDS pitch (dwords): stride%64==4 -> conflict-free 8B column reads

typedef __attribute__((ext_vector_type(2))) float v2f;
typedef __attribute__((ext_vector_type(8))) float v8f;

// ---------------- degree histogram (int) ----------------
__global__ __launch_bounds__(256) void k_count(const int* __restrict__ dst,
                                               int* __restrict__ cnt, int E) {
    int e = blockIdx.x * blockDim.x + threadIdx.x;
    if (e < E) atomicAdd(&cnt[dst[e]], 1);
}

// inv[i] = 1/max(cnt[i],1)
__global__ __launch_bounds__(256) void k_recip(const int* __restrict__ cnt,
                                               float* __restrict__ inv, int n) {
    int i = blockIdx.x * blockDim.x + threadIdx.x;
    if (i < n) inv[i] = 1.0f / fmaxf((float)cnt[i], 1.0f);
}

// ---------------- 2-level exclusive scan (n <= 256*256) ----------------
__global__ __launch_bounds__(256) void k_scan1(const int* __restrict__ cnt,
                                               int* __restrict__ rowstart,
                                               int* __restrict__ bsum, int n) {
    __shared__ int sh[256];
    int i = blockIdx.x * 256 + threadIdx.x;
    int v = (i < n) ? cnt[i] : 0;
    sh[threadIdx.x] = v;
    __syncthreads();
    for (int off = 1; off < 256; off <<= 1) {
        int t = (threadIdx.x >= off) ? sh[threadIdx.x - off] : 0;
        __syncthreads();
        sh[threadIdx.x] += t;
        __syncthreads();
    }
    if (i < n) rowstart[i] = sh[threadIdx.x] - v;   // exclusive
    if (threadIdx.x == 255) bsum[blockIdx.x] = sh[255];
}

__global__ __launch_bounds__(256) void k_scan2(int* __restrict__ bsum, int nb) {
    __shared__ int sh[256];
    int v = (threadIdx.x < nb) ? bsum[threadIdx.x] : 0;
    sh[threadIdx.x] = v;
    __syncthreads();
    for (int off = 1; off < 256; off <<= 1) {
        int t = (threadIdx.x >= off) ? sh[threadIdx.x - off] : 0;
        __syncthreads();
        sh[threadIdx.x] += t;
        __syncthreads();
    }
    if (threadIdx.x < nb) bsum[threadIdx.x] = sh[threadIdx.x] - v;   // exclusive block offsets
}

__global__ __launch_bounds__(256) void k_scan3(int* __restrict__ rowstart,
                                               const int* __restrict__ bsum,
                                               int* __restrict__ cursor, int n, int E) {
    int i = blockIdx.x * 256 + threadIdx.x;
    if (i < n) {
        int r = rowstart[i] + bsum[i >> 8];
        rowstart[i] = r;
        cursor[i] = r;
    }
    if (blockIdx.x == 0 && threadIdx.x == 0) rowstart[n] = E;
}

// ---------------- CSR fill: 1 int atomic per edge ----------------
__global__ __launch_bounds__(256) void k_fill(const int* __restrict__ srcidx,
                                              const int* __restrict__ dstidx,
                                              int* __restrict__ cursor,
                                              int* __restrict__ eidx, int E) {
    int e = blockIdx.x * blockDim.x + threadIdx.x;
    if (e < E) {
        int pos = atomicAdd(&cursor[dstidx[e]], 1);
        eidx[pos] = srcidx[e];
    }
}

// ---------------- CSR aggregation: one wave per destination row, no atomics ----------------
__global__ __launch_bounds__(256) void k_agg(const float* __restrict__ feat,
                                             const int* __restrict__ remap,
                                             const int* __restrict__ eidx,
                                             const int* __restrict__ rowstart,
                                             float* __restrict__ agg, int N) {
    int gid = blockIdx.x * blockDim.x + threadIdx.x;
    int n = gid >> 5;
    if (n >= N) return;
    int lane = gid & 31;
    int s0 = rowstart[n], s1 = rowstart[n + 1];
    float4 sum = {0.0f, 0.0f, 0.0f, 0.0f};
    for (int i = s0; i < s1; ++i) {
        int s = eidx[i];
        if (remap) s = remap[s];
        const float4 v = *(const float4*)(feat + (size_t)s * HC + lane * 4);
        sum.x += v.x; sum.y += v.y; sum.z += v.z; sum.w += v.w;
    }
    *(float4*)(agg + (size_t)n * HC + lane * 4) = sum;
}

// ---------------- fused SAGEConv GEMM (fp32 WMMA) ----------------
// out[n,:] = L2norm( (agg[n,:]*inv[n]) @ wl^T + b + xdst[g(n),:] @ wr^T ) [lrelu] [+resid]
// One wave handles a 16-row x 128-col output stripe: 8 accumulators of 16x16.
__global__ __launch_bounds__(256) void k_sage_gemm(
    const float* __restrict__ agg, const float* __restrict__ invcnt,
    const float* __restrict__ wl, const float* __restrict__ bias,
    const float* __restrict__ wr,
    const float* __restrict__ xdst, const int* __restrict__ xremap,
    const float* __restrict__ resid, float* __restrict__ out,
    int N, int lrelu) {
    extern __shared__ float lds[];

    const int tid = threadIdx.x;
    // preload both 128x128 weight matrices into LDS with padded pitch
    for (int idx = tid; idx < HC * HC; idx += 256) {
        int r = idx >> 7, c = idx & (HC - 1);
        lds[r * LDSP + c] = wl[idx];
        lds[HC * LDSP + r * LDSP + c] = wr[idx];
    }
    __syncthreads();

    const int wave = blockIdx.x * (blockDim.x >> 5) + (tid >> 5);
    const int rowbase = wave * 16;
    if (rowbase >= N) return;                 // wave-uniform; EXEC stays all-1s for WMMA
    const int lane = tid & 31;
    const int half = lane >> 4;               // selects K-pair {0,1} vs {2,3} for A/B frags
    const int lm = lane & 15;

    v8f acc[8] = {};

    const int arow = rowbase + lm;
    const float inv = invcnt[arow];
    int xr = arow;
    if (xremap) xr = xremap[xr];
    const float* ap0 = agg + (size_t)arow * HC + half * 2;
    const float* ap1 = xdst + (size_t)xr * HC + half * 2;

    const float* bpl[8];
    const float* bpr[8];
#pragma unroll
    for (int j = 0; j < 8; ++j) {
        const float* t = lds + (size_t)(j * 16 + lm) * LDSP + half * 2;
        bpl[j] = t;
        bpr[j] = t + HC * LDSP;
    }

    // merged K-loop over both GEMMs; all 16 B-fragment LDS loads (and both A global
    // loads) are forced to issue before the WMMA chains via a scheduling barrier, so
    // the hardware can retire them under staggered partial s_wait_dscnt while the
    // next k-step's loads hoist above the current WMMA chain.
#pragma unroll 4
    for (int k = 0; k < HC; k += 4) {
        v2f a0 = *(const v2f*)(ap0 + k);
        a0.x *= inv; a0.y *= inv;
        v2f a1 = *(const v2f*)(ap1 + k);
        v2f bl[8], br[8];
#pragma unroll
        for (int j = 0; j < 8; ++j) bl[j] = *(const v2f*)(bpl[j] + k);
#pragma unroll
        for (int j = 0; j < 8; ++j) br[j] = *(const v2f*)(bpr[j] + k);
        __builtin_amdgcn_sched_barrier(0);    // loads above, WMMAs below
#pragma unroll
        for (int j = 0; j < 8; ++j)
            acc[j] = __builtin_amdgcn_wmma_f32_16x16x4_f32(
                false, a0, false, bl[j], (short)0, acc[j], false, false);
#pragma unroll
        for (int j = 0; j < 8; ++j)
            acc[j] = __builtin_amdgcn_wmma_f32_16x16x4_f32(
                false, a1, false, br[j], (short)0, acc[j], false, false);
    }

    // ---- epilogue: bias, per-row L2 normalize, leaky-relu / residual, store ----
#pragma unroll
    for (int j = 0; j < 8; ++j) {
        const float bj = bias[j * 16 + lm];   // col = j*16 + lm
#pragma unroll
        for (int i = 0; i < 8; ++i) acc[j][i] += bj;
    }
#pragma unroll
    for (int i = 0; i < 8; ++i) {
        // row held in vgpr i is M = i + 8*half; its 16 N-values live across the 16-lane group
        float ss = 0.0f;
#pragma unroll
        for (int j = 0; j < 8; ++j) ss += acc[j][i] * acc[j][i];
        ss += __shfl_xor(ss, 1, 32);
        ss += __shfl_xor(ss, 2, 32);
        ss += __shfl_xor(ss, 4, 32);
        ss += __shfl_xor(ss, 8, 32);
        const float scale = 1.0f / fmaxf(sqrtf(ss), 1e-12f);
        const int row = rowbase + i + half * 8;
#pragma unroll
        for (int j = 0; j < 8; ++j) {
            float v = acc[j][i] * scale;
            if (lrelu) v = (v > 0.0f) ? v : 0.01f * v;
            const size_t addr = (size_t)row * HC + j * 16 + lm;
            if (resid) v += resid[addr];
            out[addr] = v;
        }
    }
}

// ---------------- classifier: one wave per label edge ----------------
__global__ __launch_bounds__(256) void k_classifier(const float* __restrict__ h3s,
                                                    const float* __restrict__ h3t,
                                                    const int* __restrict__ lsi,
                                                    const int* __restrict__ lti,
                                                    float* __restrict__ out, int L) {
    int gid = blockIdx.x * blockDim.x + threadIdx.x;
    int e = gid >> 5;
    if (e >= L) return;
    int lane = gid & 31;
    int s = lsi[e], t = lti[e];
    const float4 a = *(const float4*)(h3s + (size_t)s * HC + lane * 4);
    const float4 b = *(const float4*)(h3t + (size_t)t * HC + lane * 4);
    float d = a.x * b.x + a.y * b.y + a.z * b.z + a.w * b.w;
    d += __shfl_xor(d, 16, 32);
    d += __shfl_xor(d, 8, 32);
    d += __shfl_xor(d, 4, 32);
    d += __shfl_xor(d, 2, 32);
    d += __shfl_xor(d, 1, 32);
    if (lane == 0) out[e] = d;
}

extern "C" void kernel_launch(void* const* d_in, const int* in_sizes, int n_in,
                              void* d_out, int out_size, void* d_ws, size_t ws_size,
                              hipStream_t stream) {
    // inputs in setup_inputs() dict order
    const int* src_node_id = (const int*)d_in[0];
    const int* tgt_node_id = (const int*)d_in[1];
    const int* edge_index  = (const int*)d_in[2];
    const int* label_index = (const int*)d_in[3];
    const float* src_emb   = (const float*)d_in[4];
    const float* tgt_emb   = (const float*)d_in[5];
    const float* w1l_bt = (const float*)d_in[6];
    const float* w1r_bt = (const float*)d_in[7];
    const float* w1l_tb = (const float*)d_in[8];
    const float* w1r_tb = (const float*)d_in[9];
    const float* w2l_bt = (const float*)d_in[10];
    const float* w2r_bt = (const float*)d_in[11];
    const float* w2l_tb = (const float*)d_in[12];
    const float* w2r_tb = (const float*)d_in[13];
    const float* b1_bt = (const float*)d_in[14];
    const float* b1_tb = (const float*)d_in[15];
    const float* b2_bt = (const float*)d_in[16];
    const float* b2_tb = (const float*)d_in[17];

    const int NS = in_sizes[0];
    const int NT = in_sizes[1];
    const int E  = in_sizes[2] / 2;
    const int L  = in_sizes[3] / 2;
    const int NMAX = (NS > NT) ? NS : NT;

    const int* si = edge_index;          // [0,:] sources
    const int* ti = edge_index + E;      // [1,:] targets
    const int* lsi = label_index;
    const int* lti = label_index + L;

    // workspace layout: feature buffers first (16B aligned), then int/small arrays
    float* ws = (float*)d_ws;
    float* agg   = ws;                               // NMAX*HC
    float* h1_s  = agg  + (size_t)NMAX * HC;         // NS*HC
    float* h1_t  = h1_s + (size_t)NS * HC;           // NT*HC
    float* h3_s  = h1_t + (size_t)NT * HC;           // NS*HC
    float* h3_t  = h3_s + (size_t)NS * HC;           // NT*HC
    float* inv_t = h3_t + (size_t)NT * HC;           // NT
    float* inv_s = inv_t + NT;                       // NS
    int* cnti_t  = (int*)(inv_s + NS);               // NT  (zeroed together with cnti_s)
    int* cnti_s  = cnti_t + NT;                      // NS
    int* rs_t    = cnti_s + NS;                      // NT+1
    int* rs_s    = rs_t + NT + 1;                    // NS+1
    int* cur_t   = rs_s + NS + 1;                    // NT
    int* cur_s   = cur_t + NT;                       // NS
    int* bsum    = cur_s + NS;                       // 256
    int* eidx_t  = bsum + 256;                       // E
    int* eidx_s  = eidx_t + E;                       // E

    const size_t ldsBytes = 2u * HC * LDSP * sizeof(float);   // 135168 B (<320KB/WGP)
    const int blkE  = (E + 255) / 256;
    const int nbT   = (NT + 255) / 256;
    const int nbS   = (NS + 255) / 256;
    auto waveBlocks = [](int n) { return ((n << 5) + 255) / 256; };
    auto gemmGrid   = [](int N) { int tiles = (N + 15) / 16; return (tiles + 7) / 8; };

    // ---- degree histograms + reciprocals ----
    hipMemsetAsync(cnti_t, 0, (size_t)(NT + NS) * sizeof(int), stream);
    k_count<<<blkE, 256, 0, stream>>>(ti, cnti_t, E);
    k_count<<<blkE, 256, 0, stream>>>(si, cnti_s, E);
    k_recip<<<nbT, 256, 0, stream>>>(cnti_t, inv_t, NT);
    k_recip<<<nbS, 256, 0, stream>>>(cnti_s, inv_s, NS);

    // ---- build CSR once per direction (layer-invariant) ----
    k_scan1<<<nbT, 256, 0, stream>>>(cnti_t, rs_t, bsum, NT);
    k_scan2<<<1, 256, 0, stream>>>(bsum, nbT);
    k_scan3<<<nbT, 256, 0, stream>>>(rs_t, bsum, cur_t, NT, E);
    k_fill<<<blkE, 256, 0, stream>>>(si, ti, cur_t, eidx_t, E);

    k_scan1<<<nbS, 256, 0, stream>>>(cnti_s, rs_s, bsum, NS);
    k_scan2<<<1, 256, 0, stream>>>(bsum, nbS);
    k_scan3<<<nbS, 256, 0, stream>>>(rs_s, bsum, cur_s, NS, E);
    k_fill<<<blkE, 256, 0, stream>>>(ti, si, cur_s, eidx_s, E);

    // ---- layer 1, source->target ----
    k_agg<<<waveBlocks(NT), 256, 0, stream>>>(src_emb, src_node_id, eidx_t, rs_t, agg, NT);
    k_sage_gemm<<<gemmGrid(NT), 256, ldsBytes, stream>>>(
        agg, inv_t, w1l_bt, b1_bt, w1r_bt, tgt_emb, tgt_node_id, nullptr, h1_t, NT, 1);

    // ---- layer 1, target->source ----
    k_agg<<<waveBlocks(NS), 256, 0, stream>>>(tgt_emb, tgt_node_id, eidx_s, rs_s, agg, NS);
    k_sage_gemm<<<gemmGrid(NS), 256, ldsBytes, stream>>>(
        agg, inv_s, w1l_tb, b1_tb, w1r_tb, src_emb, src_node_id, nullptr, h1_s, NS, 1);

    // ---- layer 2, source->target (messages from h1_s), residual h1_t ----
    k_agg<<<waveBlocks(NT), 256, 0, stream>>>(h1_s, nullptr, eidx_t, rs_t, agg, NT);
    k_sage_gemm<<<gemmGrid(NT), 256, ldsBytes, stream>>>(
        agg, inv_t, w2l_bt, b2_bt, w2r_bt, h1_t, nullptr, h1_t, h3_t, NT, 0);

    // ---- layer 2, target->source (messages from h1_t), residual h1_s ----
    k_agg<<<waveBlocks(NS), 256, 0, stream>>>(h1_t, nullptr, eidx_s, rs_s, agg, NS);
    k_sage_gemm<<<gemmGrid(NS), 256, ldsBytes, stream>>>(
        agg, inv_s, w2l_tb, b2_tb, w2r_tb, h1_s, nullptr, h1_s, h3_s, NS, 0);

    // ---- classifier ----
    k_classifier<<<waveBlocks(L), 256, 0, stream>>>(h3_s, h3_t, lsi, lti, (float*)d_out, L);
}